// LayerwiseLinearUplift_65773129171679
// MI455X (gfx1250) — compile-verified
//
#include <hip/hip_runtime.h>
#include <hip/hip_bf16.h>

typedef __attribute__((ext_vector_type(2))) float v2f;
typedef __attribute__((ext_vector_type(8))) float v8f;

constexpr int kB    = 4096;   // batch rows
constexpr int kH    = 1024;   // hidden dim
constexpr int kL    = 16;     // layers
constexpr int kMT   = 16;     // rows per M tile (WMMA M)
constexpr int kMaxT = kB / kMT + kL;   // 272 worst-case tiles
constexpr int kS    = 4;      // N sub-tiles per wave (16 cols each) -> 64 cols/wave
constexpr int kWaves = 4;     // waves per block -> 256 cols/block

// Workspace layout:
//  ints:
//  [0..15]   counts        [16..31] cursors       [32..48] offsets(17)
//  [49]      numTiles
//  [64 .. +kMaxT)              tileLayer
//  [64+kMaxT .. +2*kMaxT)      tileStart  (index into perm)
//  [64+2*kMaxT .. +3*kMaxT)    tileValid  (rows valid in tile, <=16)
//  [64+3*kMaxT .. +kB)         perm       (row indices sorted by layer)
//  then: kH floats of zeros (safe A source for padded rows)
constexpr int WS_COUNTS  = 0;
constexpr int WS_CURSORS = 16;
constexpr int WS_OFFSETS = 32;
constexpr int WS_NTILES  = 49;
constexpr int WS_TLAYER  = 64;
constexpr int WS_TSTART  = WS_TLAYER + kMaxT;
constexpr int WS_TVALID  = WS_TSTART + kMaxT;
constexpr int WS_PERM    = WS_TVALID + kMaxT;
constexpr int WS_ZERO    = WS_PERM + kB;       // int index; reinterpret as float*

__global__ void llu_zero_kernel(int* ws) {
    int i = threadIdx.x;                       // 256 threads
    if (i < 32) ws[i] = 0;                     // counts + cursors
    float* zb = (float*)(ws + WS_ZERO);
#pragma unroll
    for (int j = 0; j < kH / 256; ++j)         // 4 KB zero region
        zb[i + 256 * j] = 0.0f;
}

__global__ void llu_count_kernel(const int* __restrict__ lid, int* ws) {
    int b = blockIdx.x * blockDim.x + threadIdx.x;
    if (b < kB) atomicAdd(&ws[WS_COUNTS + lid[b]], 1);
}

__global__ void llu_scan_kernel(int* ws) {
    if (threadIdx.x != 0) return;              // trivial serial scan over 16 buckets
    int off = 0, t = 0;
    for (int l = 0; l < kL; ++l) {
        ws[WS_OFFSETS + l] = off;
        int cnt = ws[WS_COUNTS + l];
        int nt  = (cnt + kMT - 1) / kMT;
        for (int i = 0; i < nt; ++i) {
            ws[WS_TLAYER + t] = l;
            ws[WS_TSTART + t] = off + i * kMT;
            int v = cnt - i * kMT;
            ws[WS_TVALID + t] = v < kMT ? v : kMT;
            ++t;
        }
        off += cnt;
    }
    ws[WS_OFFSETS + kL] = off;
    ws[WS_NTILES] = t;
}

__global__ void llu_scatter_kernel(const int* __restrict__ lid, int* ws) {
    int b = blockIdx.x * blockDim.x + threadIdx.x;
    if (b >= kB) return;
    int l = lid[b];
    int pos = ws[WS_OFFSETS + l] + atomicAdd(&ws[WS_CURSORS + l], 1);
    ws[WS_PERM + pos] = b;
}

// One wave computes a 16x64 fp32 output tile: 4 accumulators share one A
// fragment per k-step (1 A-load + 4 B-loads per 4 WMMAs). K loops over H in
// steps of 4 with V_WMMA_F32_16X16X4_F32. 4 waves/block -> 16x256 per block.
__global__ __launch_bounds__(128) void llu_gemm_kernel(
    const float* __restrict__ z, const float* __restrict__ w,
    const float* __restrict__ bias, const int* __restrict__ ws,
    float* __restrict__ out)
{
    const int numTiles = ws[WS_NTILES];
    const int tile = blockIdx.y;
    if (tile >= numTiles) return;

    const int l      = ws[WS_TLAYER + tile];
    const int rstart = ws[WS_TSTART + tile];
    const int nvalid = ws[WS_TVALID + tile];
    const int* perm  = ws + WS_PERM;
    const float* zeroRow = (const float*)(ws + WS_ZERO);

    const int lane = threadIdx.x & 31;
    const int wave = threadIdx.x >> 5;
    const int half = lane >> 4;          // upper half-wave holds upper K pair
    const int m    = lane & 15;          // A-row index for this lane (and N offset)
    const int n0   = blockIdx.x * (kWaves * kS * 16) + wave * (kS * 16) + m;

    // Padded rows read from the zero scratch row: no masking in the K loop.
    const bool mOk = (m < nvalid);
    const int rowA = mOk ? perm[rstart + m] : 0;
    const float* __restrict__ zrow  = mOk ? (z + (size_t)rowA * kH) : zeroRow;
    // B[k][n] = W[l][n][k]; the 4 sub-tiles sit at +s*16*kH floats (64KB, imm24-OK)
    const float* __restrict__ wbase = w + ((size_t)l * kH + n0) * kH;

    v8f acc[kS] = {};
#pragma unroll 4
    for (int k0 = 0; k0 < kH; k0 += 4) {
        const int kb = k0 + half * 2;     // ISA 32-bit A layout: v0=K{0,2}, v1=K{1,3}
        v2f a = *(const v2f*)(zrow + kb);
#pragma unroll
        for (int s = 0; s < kS; ++s) {
            v2f b = *(const v2f*)(wbase + (size_t)s * 16 * kH + kb);
            // 8 args: (neg_a, A, neg_b, B, c_mod, C, reuse_a, reuse_b)
            acc[s] = __builtin_amdgcn_wmma_f32_16x16x4_f32(
                         false, a, false, b, (short)0, acc[s], false, false);
        }
    }

    float bv[kS];
#pragma unroll
    for (int s = 0; s < kS; ++s) bv[s] = bias[l * kH + n0 + 16 * s];

#pragma unroll
    for (int i = 0; i < 8; ++i) {
        const int mr = i + half * 8;      // C layout: VGPR i -> M=i / M=i+8
        if (mr < nvalid) {
            const int row = perm[rstart + mr];
            float* __restrict__ orow = out + (size_t)row * kH + n0;
#pragma unroll
            for (int s = 0; s < kS; ++s)
                orow[16 * s] = acc[s][i] + bv[s];
        }
    }
}

extern "C" void kernel_launch(void* const* d_in, const int* in_sizes, int n_in,
                              void* d_out, int out_size, void* d_ws, size_t ws_size,
                              hipStream_t stream) {
    const float* z    = (const float*)d_in[0];
    const int*   lid  = (const int*)d_in[1];
    const float* w    = (const float*)d_in[2];
    const float* bias = (const float*)d_in[3];
    float* out = (float*)d_out;
    int*   ws  = (int*)d_ws;

    llu_zero_kernel<<<1, 256, 0, stream>>>(ws);
    llu_count_kernel<<<kB / 256, 256, 0, stream>>>(lid, ws);
    llu_scan_kernel<<<1, 32, 0, stream>>>(ws);
    llu_scatter_kernel<<<kB / 256, 256, 0, stream>>>(lid, ws);

    dim3 grid(kH / (kWaves * kS * 16), kMaxT);   // (4, 272)
    llu_gemm_kernel<<<grid, kWaves * 32, 0, stream>>>(z, w, bias, ws, out);
}